// GraphEncoder_46437186404434
// MI455X (gfx1250) — compile-verified
//
#include <hip/hip_runtime.h>
#include <math.h>

#define NNODES 50000
#define NEDGES 800000
#define NGRAPH 50
#define IN_DIM 13
#define KPAD   16
#define DTOT   64
#define HEADS  4
#define DH     16
#define LN_EPS 1e-5f

typedef __attribute__((ext_vector_type(2))) float v2f;
typedef __attribute__((ext_vector_type(8))) float v8f;

// ---------------- float <-> orderable uint (for atomic segment-max) --------
__device__ __forceinline__ unsigned fenc(float f) {
    unsigned u = __float_as_uint(f);
    return (u & 0x80000000u) ? ~u : (u | 0x80000000u);
}
__device__ __forceinline__ float fdec(unsigned e) {
    return (e & 0x80000000u) ? __uint_as_float(e ^ 0x80000000u)
                             : __uint_as_float(~e);
}
#define ENC_NEG_INF 0x007FFFFFu   // fenc(-inf)

// ---------------- generic fill ---------------------------------------------
__global__ void fill_u32(unsigned* __restrict__ p, unsigned v, int n) {
    int i = blockIdx.x * blockDim.x + threadIdx.x;
    if (i < n) p[i] = v;
}

// ---------------- zero-pad X: [N x 13] -> [N x 16] --------------------------
__global__ void pad_x(const float* __restrict__ X, float* __restrict__ Xp) {
    int i = blockIdx.x * blockDim.x + threadIdx.x;     // N*16
    if (i >= NNODES * KPAD) return;
    const int n = i >> 4, k = i & 15;
    Xp[i] = (k < IN_DIM) ? X[n * IN_DIM + k] : 0.0f;
}

// ---------------- zero-pad 4 weight mats: 13x64 -> [4][16][64] --------------
__global__ void pad_w(const float* __restrict__ Wq, const float* __restrict__ Wk,
                      const float* __restrict__ Wv, const float* __restrict__ Wr,
                      float* __restrict__ Wp) {
    int i = blockIdx.x * blockDim.x + threadIdx.x;     // 4*16*64
    if (i >= 4 * KPAD * DTOT) return;
    const int mat = i >> 10, rem = i & 1023;
    const int k = rem >> 6, c = rem & 63;
    const float* W = (mat == 0) ? Wq : (mat == 1) ? Wk : (mat == 2) ? Wv : Wr;
    Wp[i] = (k < IN_DIM) ? W[k * DTOT + c] : 0.0f;
}

// ---------------- Layer 1 projection: [N x 16p] @ [16p x 64] (x4), WMMA -----
// One wave per (rowTile, mat). 16x64 output tile: 4 accumulators, A reused.
__global__ void proj1_wmma(const float* __restrict__ Xp, const float* __restrict__ Wp,
                           float* __restrict__ Qb, float* __restrict__ Kb,
                           float* __restrict__ Vb, float* __restrict__ Rb) {
    const int lane    = threadIdx.x;
    const int rowTile = blockIdx.x >> 2;
    const int mat     = blockIdx.x & 3;
    const float* W = Wp + mat * (KPAD * DTOT);
    float*     Out = (mat == 0) ? Qb : (mat == 1) ? Kb : (mat == 2) ? Vb : Rb;

    const int half = lane >> 4;
    const int lr   = lane & 15;
    const int row  = rowTile * 16 + lr;

    v8f c[4] = {v8f{}, v8f{}, v8f{}, v8f{}};
#pragma unroll
    for (int kt = 0; kt < 4; ++kt) {
        const int k = kt * 4 + 2 * half;
        const v2f a = *(const v2f*)&Xp[row * KPAD + k];          // 8B aligned
#pragma unroll
        for (int ct = 0; ct < 4; ++ct) {
            v2f b;
            b.x = W[ k      * DTOT + ct * 16 + lr];
            b.y = W[(k + 1) * DTOT + ct * 16 + lr];
            c[ct] = __builtin_amdgcn_wmma_f32_16x16x4_f32(false, a, false, b,
                                                          (short)0, c[ct], false, false);
        }
    }
#pragma unroll
    for (int ct = 0; ct < 4; ++ct)
#pragma unroll
        for (int v = 0; v < 8; ++v) {
            const int r = rowTile * 16 + v + 8 * half;
            Out[r * DTOT + ct * 16 + lr] = c[ct][v];
        }
}

// ---------------- Layer 2 projection: [N x 64] @ 64x64 (x3), WMMA -----------
__global__ void proj2_wmma(const float* __restrict__ H,
                           const float* __restrict__ Wq, const float* __restrict__ Wk,
                           const float* __restrict__ Wv,
                           float* __restrict__ Qb, float* __restrict__ Kb,
                           float* __restrict__ Vb) {
    const int lane    = threadIdx.x;
    const int rowTile = blockIdx.x / 3;
    const int mat     = blockIdx.x - rowTile * 3;
    const float* W = (mat == 0) ? Wq : (mat == 1) ? Wk : Wv;
    float*     Out = (mat == 0) ? Qb : (mat == 1) ? Kb : Vb;

    const int half = lane >> 4;
    const int lr   = lane & 15;
    const int row  = rowTile * 16 + lr;

    v8f c[4] = {v8f{}, v8f{}, v8f{}, v8f{}};
#pragma unroll
    for (int kt = 0; kt < 16; ++kt) {
        const int k = kt * 4 + 2 * half;
        const v2f a = *(const v2f*)&H[row * DTOT + k];           // 8B aligned
#pragma unroll
        for (int ct = 0; ct < 4; ++ct) {
            v2f b;
            b.x = W[ k      * DTOT + ct * 16 + lr];
            b.y = W[(k + 1) * DTOT + ct * 16 + lr];
            c[ct] = __builtin_amdgcn_wmma_f32_16x16x4_f32(false, a, false, b,
                                                          (short)0, c[ct], false, false);
        }
    }
#pragma unroll
    for (int ct = 0; ct < 4; ++ct)
#pragma unroll
        for (int v = 0; v < 8; ++v) {
            const int r = rowTile * 16 + v + 8 * half;
            Out[r * DTOT + ct * 16 + lr] = c[ct][v];
        }
}

// ---------------- edge score helper (float4 loads) --------------------------
__device__ __forceinline__ float edge_score(const float* __restrict__ q,
                                            const float* __restrict__ k,
                                            int dst, int src, int h) {
    const float4* qp = (const float4*)(q + dst * DTOT + h * DH);   // 16B aligned
    const float4* kp = (const float4*)(k + src * DTOT + h * DH);
    float s = 0.0f;
#pragma unroll
    for (int j = 0; j < 4; ++j) {
        const float4 a = qp[j], b = kp[j];
        s = fmaf(a.x, b.x, s); s = fmaf(a.y, b.y, s);
        s = fmaf(a.z, b.z, s); s = fmaf(a.w, b.w, s);
    }
    return s * 0.25f;                                  // 1/sqrt(16)
}

// ---------------- pass 1: segment max ---------------------------------------
__global__ void edge_max(const float* __restrict__ q, const float* __restrict__ k,
                         const int* __restrict__ src, const int* __restrict__ dst,
                         unsigned* __restrict__ mEnc, int EH) {
    int i = blockIdx.x * blockDim.x + threadIdx.x;
    if (i >= EH) return;
    const int e = i >> 2, h = i & 3;
    const float s = edge_score(q, k, dst[e], src[e], h);
    atomicMax(&mEnc[dst[e] * HEADS + h], fenc(s));
}

// decode max (and zero nodes with no in-edges, matching isfinite guard)
__global__ void m_finalize(unsigned* __restrict__ mEnc, float* __restrict__ mOut, int n) {
    int i = blockIdx.x * blockDim.x + threadIdx.x;
    if (i >= n) return;
    const unsigned e = mEnc[i];
    mOut[i] = (e == ENC_NEG_INF) ? 0.0f : fdec(e);
}

// ---------------- pass 2: segment sum of exp --------------------------------
__global__ void edge_denom(const float* __restrict__ q, const float* __restrict__ k,
                           const float* __restrict__ m,
                           const int* __restrict__ src, const int* __restrict__ dst,
                           float* __restrict__ denom, int EH) {
    int i = blockIdx.x * blockDim.x + threadIdx.x;
    if (i >= EH) return;
    const int e = i >> 2, h = i & 3;
    const int d = dst[e];
    const float s  = edge_score(q, k, d, src[e], h);
    const float ex = __expf(s - m[d * HEADS + h]);
    atomicAdd(&denom[d * HEADS + h], ex);
}

// ---------------- pass 3: weighted aggregation ------------------------------
__global__ void edge_agg(const float* __restrict__ q, const float* __restrict__ k,
                         const float* __restrict__ v, const float* __restrict__ m,
                         const float* __restrict__ denom,
                         const int* __restrict__ src, const int* __restrict__ dst,
                         float* __restrict__ agg, int EH) {
    int i = blockIdx.x * blockDim.x + threadIdx.x;
    if (i >= EH) return;
    const int e = i >> 2, h = i & 3;
    const int dn = dst[e], sn = src[e];
    const float s = edge_score(q, k, dn, sn, h);
    const float w = __expf(s - m[dn * HEADS + h]) / (denom[dn * HEADS + h] + 1e-16f);
    const float4* vp = (const float4*)(v + sn * DTOT + h * DH);
    float*        ap = agg + dn * DTOT + h * DH;
#pragma unroll
    for (int j = 0; j < 4; ++j) {
        const float4 vv = vp[j];
        atomicAdd(&ap[j * 4 + 0], w * vv.x);
        atomicAdd(&ap[j * 4 + 1], w * vv.y);
        atomicAdd(&ap[j * 4 + 2], w * vv.z);
        atomicAdd(&ap[j * 4 + 3], w * vv.w);
    }
}

// ---------------- residual + LayerNorm (64 threads / node) ------------------
__global__ void add_res_ln(const float* __restrict__ agg, const float* __restrict__ res,
                           const float* __restrict__ g, const float* __restrict__ b,
                           float* __restrict__ out) {
    __shared__ float sh[DTOT];
    const int n = blockIdx.x, c = threadIdx.x;
    const float h = agg[n * DTOT + c] + res[n * DTOT + c];
    sh[c] = h; __syncthreads();
    for (int s = 32; s > 0; s >>= 1) { if (c < s) sh[c] += sh[c + s]; __syncthreads(); }
    const float mu = sh[0] * (1.0f / DTOT);
    __syncthreads();
    const float d = h - mu;
    sh[c] = d * d; __syncthreads();
    for (int s = 32; s > 0; s >>= 1) { if (c < s) sh[c] += sh[c + s]; __syncthreads(); }
    const float var = sh[0] * (1.0f / DTOT);
    out[n * DTOT + c] = g[c] * d * rsqrtf(var + LN_EPS) + b[c];
}

// ---------------- per-graph mean pooling ------------------------------------
__global__ void pool_mean(const float* __restrict__ H2, const int* __restrict__ ptr,
                          float* __restrict__ Q) {
    const int gph = blockIdx.x, c = threadIdx.x;
    const int s = ptr[gph], e = ptr[gph + 1];
    float acc = 0.0f;
    for (int n = s; n < e; ++n) acc += H2[n * DTOT + c];
    Q[gph * DTOT + c] = acc / (float)(e - s);
}

// ===========================================================================
extern "C" void kernel_launch(void* const* d_in, const int* in_sizes, int n_in,
                              void* d_out, int out_size, void* d_ws, size_t ws_size,
                              hipStream_t stream) {
    const float* nodes = (const float*)d_in[0];
    const int*   eidx  = (const int*)d_in[1];
    const int*   bptr  = (const int*)d_in[2];
    const float* Wq1 = (const float*)d_in[3];
    const float* Wk1 = (const float*)d_in[4];
    const float* Wv1 = (const float*)d_in[5];
    const float* Wr1 = (const float*)d_in[6];
    const float* g1  = (const float*)d_in[7];
    const float* b1  = (const float*)d_in[8];
    const float* Wq2 = (const float*)d_in[9];
    const float* Wk2 = (const float*)d_in[10];
    const float* Wv2 = (const float*)d_in[11];
    const float* g2  = (const float*)d_in[12];
    const float* b2  = (const float*)d_in[13];

    const int* esrc = eidx;            // edge_index[0, :]
    const int* edst = eidx + NEDGES;   // edge_index[1, :]

    // workspace layout (floats)
    const size_t NH = (size_t)NNODES * DTOT;          // 3.2M
    float* qbuf  = (float*)d_ws;
    float* kbuf  = qbuf + NH;
    float* vbuf  = kbuf + NH;
    float* rbuf  = vbuf + NH;
    float* agg   = rbuf + NH;
    float* H1    = agg  + NH;
    float* mbuf  = H1   + NH;                          // N*4 floats (aliased uint)
    float* denom = mbuf + (size_t)NNODES * HEADS;
    float* Xp    = denom + (size_t)NNODES * HEADS;     // N*16
    float* Wp    = Xp + (size_t)NNODES * KPAD;         // 4*16*64

    float* H2 = (float*)d_out;                         // N*64
    float* Q  = H2 + NH;                               // 50*64

    const int rowTiles = NNODES / 16;                  // 3125
    const int EH       = NEDGES * HEADS;               // 3.2M
    const int NHh      = NNODES * HEADS;               // 200K
    const int TB = 256;
    const dim3 eg((EH + TB - 1) / TB), eb(TB);
    const dim3 sg((NHh + TB - 1) / TB);
    const dim3 ag(((int)NH + TB - 1) / TB);

    // ---------------- padded copies (layer 1) ----------------
    pad_x<<<(NNODES * KPAD + TB - 1) / TB, TB, 0, stream>>>(nodes, Xp);
    pad_w<<<(4 * KPAD * DTOT + TB - 1) / TB, TB, 0, stream>>>(Wq1, Wk1, Wv1, Wr1, Wp);

    // ---------------- Layer 1 ----------------
    proj1_wmma<<<rowTiles * 4, 32, 0, stream>>>(Xp, Wp, qbuf, kbuf, vbuf, rbuf);
    fill_u32<<<sg, eb, 0, stream>>>((unsigned*)mbuf, ENC_NEG_INF, NHh);
    fill_u32<<<sg, eb, 0, stream>>>((unsigned*)denom, 0u, NHh);
    fill_u32<<<ag, eb, 0, stream>>>((unsigned*)agg, 0u, (int)NH);

    edge_max  <<<eg, eb, 0, stream>>>(qbuf, kbuf, esrc, edst, (unsigned*)mbuf, EH);
    m_finalize<<<sg, eb, 0, stream>>>((unsigned*)mbuf, mbuf, NHh);
    edge_denom<<<eg, eb, 0, stream>>>(qbuf, kbuf, mbuf, esrc, edst, denom, EH);
    edge_agg  <<<eg, eb, 0, stream>>>(qbuf, kbuf, vbuf, mbuf, denom, esrc, edst, agg, EH);
    add_res_ln<<<NNODES, DTOT, 0, stream>>>(agg, rbuf, g1, b1, H1);

    // ---------------- Layer 2 ----------------
    proj2_wmma<<<rowTiles * 3, 32, 0, stream>>>(H1, Wq2, Wk2, Wv2, qbuf, kbuf, vbuf);
    fill_u32<<<sg, eb, 0, stream>>>((unsigned*)mbuf, ENC_NEG_INF, NHh);
    fill_u32<<<sg, eb, 0, stream>>>((unsigned*)denom, 0u, NHh);
    fill_u32<<<ag, eb, 0, stream>>>((unsigned*)agg, 0u, (int)NH);

    edge_max  <<<eg, eb, 0, stream>>>(qbuf, kbuf, esrc, edst, (unsigned*)mbuf, EH);
    m_finalize<<<sg, eb, 0, stream>>>((unsigned*)mbuf, mbuf, NHh);
    edge_denom<<<eg, eb, 0, stream>>>(qbuf, kbuf, mbuf, esrc, edst, denom, EH);
    edge_agg  <<<eg, eb, 0, stream>>>(qbuf, kbuf, vbuf, mbuf, denom, esrc, edst, agg, EH);
    add_res_ln<<<NNODES, DTOT, 0, stream>>>(agg, H1, g2, b2, H2);

    // ---------------- Pooling ----------------
    pool_mean<<<NGRAPH, DTOT, 0, stream>>>(H2, bptr, Q);
}